// BlockUncertaintyTracker_45775761441260
// MI455X (gfx1250) — compile-verified
//
#include <hip/hip_runtime.h>

// Problem constants (match reference: B=16, C=1, H=W=2048, BS=4)
#define WIDTH   2048
#define HEIGHT  2048
#define BATCH   16
#define IMG     (WIDTH * HEIGHT)      // 4 M elements per batch image
#define NBW     512                   // blocks per row (W/4)
#define TPB     256

typedef float v4f __attribute__((ext_vector_type(4)));
typedef __attribute__((address_space(3))) float* lds_fptr;

// CDNA5 async copy: global -> LDS, 128b per lane, tracked on ASYNCcnt.
__device__ __forceinline__ void async_copy_b128(lds_fptr lds_dst, const float* gsrc) {
    asm volatile("global_load_async_to_lds_b128 %0, %1, off"
                 :: "v"(lds_dst), "v"(gsrc)
                 : "memory");
}

__global__ __launch_bounds__(TPB) void block_uncertainty_kernel(
    const float* __restrict__ cur,      // [16,1,2048,2048]
    const float* __restrict__ ema_err,  // [262144]
    const float* __restrict__ ema_q,    // [262144]
    float* __restrict__ out)            // [16,1,2048,2048]
{
    // [stage][row][tid][4] : each thread's 4x4 block staged per batch, 32 KB
    __shared__ float smem[2][4][TPB][4];

    const int tid = threadIdx.x;
    const int nb  = blockIdx.x * TPB + tid;       // block index, 0..262143
    const int bi  = nb >> 9;                      // nb / NBW
    const int bj  = nb & (NBW - 1);               // nb % NBW
    const size_t base = (size_t)(bi * 4) * WIDTH + (size_t)(bj * 4);

    // ---- prefetch batch 0 into stage 0 ----
    #pragma unroll
    for (int r = 0; r < 4; ++r) {
        async_copy_b128((lds_fptr)&smem[0][r][tid][0],
                        cur + base + (size_t)r * WIDTH);
    }

    float mean_acc = 0.0f;
    float q_acc    = 0.0f;

    #pragma unroll
    for (int b = 0; b < BATCH; ++b) {
        const int stage = b & 1;

        if (b + 1 < BATCH) {
            // issue next batch's 4 async b128 loads into the other stage
            #pragma unroll
            for (int r = 0; r < 4; ++r) {
                async_copy_b128((lds_fptr)&smem[stage ^ 1][r][tid][0],
                                cur + (size_t)(b + 1) * IMG + base + (size_t)r * WIDTH);
            }
            // 8 outstanding; <=4 left means batch b's loads have landed (in-order retire)
            asm volatile("s_wait_asynccnt 0x4" ::: "memory");
        } else {
            asm volatile("s_wait_asynccnt 0x0" ::: "memory");
        }

        // per-batch block statistics: sum + top-3 (quantile 0.9 over 16 elems
        // = 0.5*(sorted[13]+sorted[14]) = 0.5*(2nd-largest + 3rd-largest))
        float m1 = -__builtin_inff();
        float m2 = -__builtin_inff();
        float m3 = -__builtin_inff();
        float sum = 0.0f;
        #pragma unroll
        for (int r = 0; r < 4; ++r) {
            v4f v = *(const v4f*)&smem[stage][r][tid][0];   // ds_load_b128
            #pragma unroll
            for (int k = 0; k < 4; ++k) {
                const float x = v[k];
                sum += x;
                const float t1 = fminf(x, m1);
                m1 = fmaxf(x, m1);
                const float t2 = fminf(t1, m2);
                m2 = fmaxf(t1, m2);
                m3 = fmaxf(t2, m3);
            }
        }
        mean_acc += sum * (1.0f / 16.0f);
        q_acc    += 0.5f * (m2 + m3);
    }

    const float block_mean = mean_acc * (1.0f / (float)BATCH);
    const float block_q    = q_acc    * (1.0f / (float)BATCH);

    const float DEC = 0.99f;
    const float OMD = (float)(1.0 - 0.99);   // match JAX: double 1-DECAY rounded to f32
    const float nerr = DEC * ema_err[nb] + OMD * block_mean;
    const float nq   = DEC * ema_q[nb]   + OMD * block_q;
    const float u    = nq / (nerr + 1e-5f);

    // broadcast u back onto the 4x4 block of every batch image, NT 128-bit stores
    const v4f uv = { u, u, u, u };
    #pragma unroll
    for (int b = 0; b < BATCH; ++b) {
        #pragma unroll
        for (int r = 0; r < 4; ++r) {
            v4f* p = (v4f*)(out + (size_t)b * IMG + base + (size_t)r * WIDTH);
            __builtin_nontemporal_store(uv, p);
        }
    }
}

extern "C" void kernel_launch(void* const* d_in, const int* in_sizes, int n_in,
                              void* d_out, int out_size, void* d_ws, size_t ws_size,
                              hipStream_t stream) {
    const float* cur      = (const float*)d_in[0];
    const float* ema_err  = (const float*)d_in[1];
    const float* ema_q    = (const float*)d_in[2];
    float*       out      = (float*)d_out;

    const int NB = in_sizes[1];           // 262144 blocks
    dim3 grid((unsigned)(NB / TPB)), block(TPB);
    hipLaunchKernelGGL(block_uncertainty_kernel, grid, block, 0, stream,
                       cur, ema_err, ema_q, out);
}